// GraphSAGE_6193342841309
// MI455X (gfx1250) — compile-verified
//
#include <hip/hip_runtime.h>

#define NNODES 100000
#define NEDGES 1600000
#define D 128
#define TSTRIDE 260   // LDS tile stride: 260 % 64 == 4 -> conflict-free b64 fragment reads
#define MTILE 64      // rows per block (4 x 16-row WMMA tiles, shared B fragments)

typedef float v2f __attribute__((ext_vector_type(2)));
typedef float v8f __attribute__((ext_vector_type(8)));

__device__ __forceinline__ void atomic_add_noret(float* p, float v) {
    // result unused -> backend emits non-returning global_atomic_add_f32 (STOREcnt path)
    (void)__hip_atomic_fetch_add(p, v, __ATOMIC_RELAXED, __HIP_MEMORY_SCOPE_AGENT);
}

__global__ void zero_kernel(float* __restrict__ p, long n) {
    long i = (long)blockIdx.x * blockDim.x + threadIdx.x;
    long stride = (long)gridDim.x * blockDim.x;
    for (; i < n; i += stride) p[i] = 0.0f;
}

__global__ void deg_kernel(const int* __restrict__ dst, float* __restrict__ deg, int E) {
    int e = blockIdx.x * blockDim.x + threadIdx.x;
    if (e < E) atomic_add_noret(&deg[dst[e]], 1.0f);
}

__global__ void invdeg_kernel(float* __restrict__ deg, int n) {
    int i = blockIdx.x * blockDim.x + threadIdx.x;
    if (i < n) deg[i] = 1.0f / fmaxf(deg[i], 1.0f);
}

// 32 lanes per edge; lane handles one float4 chunk of the 128-wide feature row.
// Raw sums only; mean scaling (1/deg) is fused into the GEMM A-tile load.
__global__ void agg_kernel(const float* __restrict__ x, const int* __restrict__ src,
                           const int* __restrict__ dst, float* __restrict__ neigh, int E) {
    long tid = (long)blockIdx.x * blockDim.x + threadIdx.x;
    int e = (int)(tid >> 5);
    int lane = (int)(tid & 31);
    if (e >= E) return;
    int s = src[e];
    int d = dst[e];
    float4 v = ((const float4*)(x + (size_t)s * D))[lane];
    float* o = neigh + (size_t)d * D + lane * 4;
    atomic_add_noret(o + 0, v.x);
    atomic_add_noret(o + 1, v.y);
    atomic_add_noret(o + 2, v.z);
    atomic_add_noret(o + 3, v.w);
}

// out[64 rows x 128 cols] = [in0 | scale*in1] (64x256) @ [Wself ; Wneigh] (256x128) + bias
// 8 waves per block, each wave owns a 16-column slice and 4 M-tiles; each B fragment
// (2 floats/lane from global) feeds 4 x V_WMMA_F32_16X16X4_F32.
template <bool RELU_DROP>
__global__ __launch_bounds__(256)
void sage_gemm_kernel(const float* __restrict__ in0,    // [N,128] self features
                      const float* __restrict__ in1,    // [N,128] raw neighbor sums
                      const float* __restrict__ scale,  // [N]     1/max(deg,1)
                      const float* __restrict__ Wself,  // [128,128] row-major
                      const float* __restrict__ Wneigh, // [128,128] row-major
                      const float* __restrict__ bias,   // [128]
                      const float* __restrict__ drop_u, // [N,128] (layer 1 only)
                      float* __restrict__ out, int n_nodes) {
    __shared__ float tileA[MTILE][TSTRIDE];

    const int t = threadIdx.x;
    const int rowbase = blockIdx.x * MTILE;

    // Cooperative load of the 64x256 concatenated A tile (4096 float4 slots).
    #pragma unroll
    for (int it = 0; it < 16; ++it) {
        int f4  = t + 256 * it;   // 0..4095
        int row = f4 >> 6;        // 64 float4 per 256-wide row
        int col = (f4 & 63) * 4;
        int node = rowbase + row;
        float4 v = make_float4(0.0f, 0.0f, 0.0f, 0.0f);
        if (node < n_nodes) {
            if (col < 128) {
                v = *(const float4*)(in0 + (size_t)node * D + col);
            } else {
                v = *(const float4*)(in1 + (size_t)node * D + (col - 128));
                float sc = scale[node];
                v.x *= sc; v.y *= sc; v.z *= sc; v.w *= sc;
            }
        }
        *(float4*)&tileA[row][col] = v;   // (row*260 + col)*4 is 16B aligned
    }
    __syncthreads();

    const int lane   = t & 31;
    const int lane16 = lane & 15;
    const int half   = lane >> 4;        // K-pair selector for A/B fragments
    const int n      = (t >> 5) * 16 + lane16;  // this lane's output column

    v8f acc0 = {}, acc1 = {}, acc2 = {}, acc3 = {};

    // K = 0..127 : x-part against W_self
    #pragma unroll 2
    for (int k = 0; k < 128; k += 4) {
        v2f b;
        b.x = Wself[(k + 2 * half) * D + n];
        b.y = Wself[(k + 2 * half + 1) * D + n];
        float2 a0 = *(const float2*)&tileA[ 0 + lane16][k + 2 * half];
        float2 a1 = *(const float2*)&tileA[16 + lane16][k + 2 * half];
        float2 a2 = *(const float2*)&tileA[32 + lane16][k + 2 * half];
        float2 a3 = *(const float2*)&tileA[48 + lane16][k + 2 * half];
        v2f av;
        av.x = a0.x; av.y = a0.y;
        acc0 = __builtin_amdgcn_wmma_f32_16x16x4_f32(false, av, false, b, (short)0, acc0, false, false);
        av.x = a1.x; av.y = a1.y;
        acc1 = __builtin_amdgcn_wmma_f32_16x16x4_f32(false, av, false, b, (short)0, acc1, false, false);
        av.x = a2.x; av.y = a2.y;
        acc2 = __builtin_amdgcn_wmma_f32_16x16x4_f32(false, av, false, b, (short)0, acc2, false, false);
        av.x = a3.x; av.y = a3.y;
        acc3 = __builtin_amdgcn_wmma_f32_16x16x4_f32(false, av, false, b, (short)0, acc3, false, false);
    }
    // K = 128..255 : neigh-part against W_neigh
    #pragma unroll 2
    for (int k = 0; k < 128; k += 4) {
        v2f b;
        b.x = Wneigh[(k + 2 * half) * D + n];
        b.y = Wneigh[(k + 2 * half + 1) * D + n];
        float2 a0 = *(const float2*)&tileA[ 0 + lane16][128 + k + 2 * half];
        float2 a1 = *(const float2*)&tileA[16 + lane16][128 + k + 2 * half];
        float2 a2 = *(const float2*)&tileA[32 + lane16][128 + k + 2 * half];
        float2 a3 = *(const float2*)&tileA[48 + lane16][128 + k + 2 * half];
        v2f av;
        av.x = a0.x; av.y = a0.y;
        acc0 = __builtin_amdgcn_wmma_f32_16x16x4_f32(false, av, false, b, (short)0, acc0, false, false);
        av.x = a1.x; av.y = a1.y;
        acc1 = __builtin_amdgcn_wmma_f32_16x16x4_f32(false, av, false, b, (short)0, acc1, false, false);
        av.x = a2.x; av.y = a2.y;
        acc2 = __builtin_amdgcn_wmma_f32_16x16x4_f32(false, av, false, b, (short)0, acc2, false, false);
        av.x = a3.x; av.y = a3.y;
        acc3 = __builtin_amdgcn_wmma_f32_16x16x4_f32(false, av, false, b, (short)0, acc3, false, false);
    }

    // Epilogue: C/D layout -> VGPR j holds M = j (lanes 0-15) / M = j+8 (lanes 16-31)
    float bcol = bias[n];
    v8f accs[4] = {acc0, acc1, acc2, acc3};
    #pragma unroll
    for (int mt = 0; mt < 4; ++mt) {
        #pragma unroll
        for (int j = 0; j < 8; ++j) {
            int node = rowbase + mt * 16 + j + 8 * half;
            if (node < n_nodes) {
                float val = accs[mt][j] + bcol;
                if (RELU_DROP) {
                    val = fmaxf(val, 0.0f);
                    val *= (drop_u[(size_t)node * D + n] > 0.5f) ? 2.0f : 0.0f;
                }
                out[(size_t)node * D + n] = val;
            }
        }
    }
}

extern "C" void kernel_launch(void* const* d_in, const int* in_sizes, int n_in,
                              void* d_out, int out_size, void* d_ws, size_t ws_size,
                              hipStream_t stream) {
    const float* x       = (const float*)d_in[0];
    const int*   src     = (const int*)d_in[1];
    const int*   dst     = (const int*)d_in[2];
    const float* Wself1  = (const float*)d_in[3];
    const float* Wneigh1 = (const float*)d_in[4];
    const float* b1      = (const float*)d_in[5];
    const float* Wself2  = (const float*)d_in[6];
    const float* Wneigh2 = (const float*)d_in[7];
    const float* b2      = (const float*)d_in[8];
    const float* dropu   = (const float*)d_in[9];
    float* out = (float*)d_out;

    const long ND = (long)NNODES * D;
    float* neigh = (float*)d_ws;                    // [N,128] neighbor sums
    float* h     = neigh + ND;                      // [N,128] layer-1 output
    float* deg   = h + ND;                          // [N]     degree -> inv_deg

    const int aggBlocks = (int)(((long)NEDGES * 32 + 255) / 256);
    const int gemmBlocks = (NNODES + MTILE - 1) / MTILE;

    zero_kernel<<<2048, 256, 0, stream>>>(neigh, ND);
    zero_kernel<<<256, 256, 0, stream>>>(deg, NNODES);
    deg_kernel<<<(NEDGES + 255) / 256, 256, 0, stream>>>(dst, deg, NEDGES);
    invdeg_kernel<<<(NNODES + 255) / 256, 256, 0, stream>>>(deg, NNODES);

    // Layer 1
    agg_kernel<<<aggBlocks, 256, 0, stream>>>(x, src, dst, neigh, NEDGES);
    sage_gemm_kernel<true><<<gemmBlocks, 256, 0, stream>>>(
        x, neigh, deg, Wself1, Wneigh1, b1, dropu, h, NNODES);

    // Layer 2
    zero_kernel<<<2048, 256, 0, stream>>>(neigh, ND);
    agg_kernel<<<aggBlocks, 256, 0, stream>>>(h, src, dst, neigh, NEDGES);
    sage_gemm_kernel<false><<<gemmBlocks, 256, 0, stream>>>(
        h, neigh, deg, Wself2, Wneigh2, b2, nullptr, out, NNODES);
}